// GCNLayer_7481833030311
// MI455X (gfx1250) — compile-verified
//
#include <hip/hip_runtime.h>
#include <hip/hip_bf16.h>

typedef __attribute__((ext_vector_type(2))) float v2f;
typedef __attribute__((ext_vector_type(8))) float v8f;

#define D_DIM 128  // D_IN == D_OUT == 128

// ---------------------------------------------------------------------------
// Kernel 1: out[n][j] = bias[j]  (scatter-add target must be initialized)
// ---------------------------------------------------------------------------
__global__ __launch_bounds__(256) void init_out_kernel(float* __restrict__ out,
                                                       const float* __restrict__ bias,
                                                       int total) {
    int i = blockIdx.x * 256 + threadIdx.x;
    if (i < total) out[i] = bias[i & (D_DIM - 1)];
}

// ---------------------------------------------------------------------------
// Kernel 2: support = x @ W^T  via V_WMMA_F32_16X16X4_F32
//   grid.x  = n_nodes/16 (M tiles), blockDim = 256 (8 waves),
//   wave w computes N-tile w (N = 128 = 8 * 16).
//   A (16x4 f32): lane L -> m = L&15, khalf = L>>4; VGPR v -> k = kb + v + 2*khalf
//     => contiguous float2 at x[(mBase+m)*128 + kb + 2*khalf]
//   B (4x16 f32): B[k][n] = W[nBase+n][k]; same per-lane float2 pattern from W row.
//   C/D (16x16 f32, 8 VGPRs): VGPR v, lanes 0-15 -> M=v, lanes 16-31 -> M=v+8.
// ---------------------------------------------------------------------------
__global__ __launch_bounds__(256) void gemm_wmma_kernel(const float* __restrict__ x,
                                                        const float* __restrict__ W,
                                                        float* __restrict__ support) {
    const int lane  = threadIdx.x & 31;
    const int wave  = threadIdx.x >> 5;          // N tile index 0..7
    const int mBase = blockIdx.x * 16;
    const int nBase = wave * 16;
    const int mn    = lane & 15;                 // row of A fragment / col of B fragment
    const int khalf = lane >> 4;                 // 0 or 1

    const float* aRow = x + (size_t)(mBase + mn) * D_DIM + 2 * khalf;
    const float* bRow = W + (size_t)(nBase + mn) * D_DIM + 2 * khalf;

    v8f c = {};
#pragma unroll
    for (int kb = 0; kb < D_DIM; kb += 4) {
        v2f a = *(const v2f*)(aRow + kb);
        v2f b = *(const v2f*)(bRow + kb);
        // (neg_a, A, neg_b, B, c_mod, C, reuse_a, reuse_b)
        c = __builtin_amdgcn_wmma_f32_16x16x4_f32(false, a, false, b, (short)0, c,
                                                  false, false);
    }

    float* outBase = support + (size_t)mBase * D_DIM + nBase + mn;
#pragma unroll
    for (int v = 0; v < 8; ++v) {
        outBase[(size_t)(v + 8 * khalf) * D_DIM] = c[v];
    }
}

// ---------------------------------------------------------------------------
// Kernel 3: edge scatter  out[row[e]] += val[e] * support[col[e]]
//   one wave (32 lanes) per edge; each lane owns 4 consecutive floats.
//   unsafeAtomicAdd -> hardware global_atomic_add_f32 (resolves at L2;
//   support+out fit in the 192MB L2, so this runs near L2 bandwidth).
// ---------------------------------------------------------------------------
__global__ __launch_bounds__(256) void edge_scatter_kernel(const int* __restrict__ edge_row,
                                                           const int* __restrict__ edge_col,
                                                           const float* __restrict__ edge_val,
                                                           const float* __restrict__ support,
                                                           float* __restrict__ out,
                                                           int n_edges) {
    const int e = blockIdx.x * 8 + (threadIdx.x >> 5);
    if (e >= n_edges) return;
    const int lane = threadIdx.x & 31;

    const int   r = edge_row[e];
    const int   c = edge_col[e];
    const float v = edge_val[e];

    const float4 m = ((const float4*)(support + (size_t)c * D_DIM))[lane];
    float* dst = out + (size_t)r * D_DIM + lane * 4;

    unsafeAtomicAdd(dst + 0, v * m.x);
    unsafeAtomicAdd(dst + 1, v * m.y);
    unsafeAtomicAdd(dst + 2, v * m.z);
    unsafeAtomicAdd(dst + 3, v * m.w);
}

// ---------------------------------------------------------------------------
// Host launcher
// Inputs (setup_inputs order): x[f32], edge_row[i32], edge_col[i32],
//                              edge_val[f32], W[f32], bias[f32]
// ---------------------------------------------------------------------------
extern "C" void kernel_launch(void* const* d_in, const int* in_sizes, int n_in,
                              void* d_out, int out_size, void* d_ws, size_t ws_size,
                              hipStream_t stream) {
    const float* x        = (const float*)d_in[0];
    const int*   edge_row = (const int*)d_in[1];
    const int*   edge_col = (const int*)d_in[2];
    const float* edge_val = (const float*)d_in[3];
    const float* W        = (const float*)d_in[4];
    const float* bias     = (const float*)d_in[5];

    float* out     = (float*)d_out;
    float* support = (float*)d_ws;   // n_nodes * 128 floats (25.6 MB)

    const int n_nodes = in_sizes[0] / D_DIM;
    const int n_edges = in_sizes[1];

    // 1) out = bias (broadcast)
    {
        int total  = out_size;
        int blocks = (total + 255) / 256;
        init_out_kernel<<<blocks, 256, 0, stream>>>(out, bias, total);
    }

    // 2) support = x @ W^T   (n_nodes is a multiple of 16: 50000 = 3125*16)
    {
        int mTiles = n_nodes / 16;
        gemm_wmma_kernel<<<mTiles, 256, 0, stream>>>(x, W, support);
    }

    // 3) scatter-add messages along edges
    {
        int blocks = (n_edges + 7) / 8;
        edge_scatter_kernel<<<blocks, 256, 0, stream>>>(edge_row, edge_col, edge_val,
                                                        support, out, n_edges);
    }
}